// Backprojection3DConsistencyLoss_42915313222175
// MI455X (gfx1250) — compile-verified
//
#include <hip/hip_runtime.h>
#include <math.h>

#define DETN   128
#define VDIM   128
#define NSAMP  512
#define NVOX   (VDIM*VDIM*VDIM)      /* 2097152 voxels */
#define NB     2                      /* batch size (reference fixes B=2) */
#define NVOLS  (NB*2)                 /* 4 binary volumes, u8, 2MB each */

typedef __attribute__((ext_vector_type(2))) float v2f;
typedef __attribute__((ext_vector_type(8))) float v8f;
typedef __attribute__((ext_vector_type(4))) int   v4i;

// Pointer flavors for the async global->LDS builtin (param 0 is AS1 v4i*).
typedef __attribute__((address_space(1))) v4i* g_v4i_p;
typedef __attribute__((address_space(3))) v4i* l_v4i_p;

// gfx1250 async global->LDS path (guarded: falls back to direct loads if the
// toolchain does not declare the async builtins).
#if defined(__AMDGCN__) && __has_builtin(__builtin_amdgcn_global_load_async_to_lds_b128) && __has_builtin(__builtin_amdgcn_s_wait_asynccnt)
#define HAVE_ASYNC_LDS 1
#else
#define HAVE_ASYNC_LDS 0
#endif

// ---------------------------------------------------------------------------
// Kernel 0: zero the 8MB of u8 volumes (as uint4) and the scalar output.
// ---------------------------------------------------------------------------
__global__ __launch_bounds__(256) void zero_init(uint4* __restrict__ w,
                                                 float* __restrict__ out, int n)
{
  int i = blockIdx.x * blockDim.x + threadIdx.x;
  if (i == 0) out[0] = 0.0f;
  uint4 z = make_uint4(0u, 0u, 0u, 0u);
  int stride = gridDim.x * blockDim.x;
  for (; i < n; i += stride) w[i] = z;
}

// ---------------------------------------------------------------------------
// Kernel 1: ray-march scatter. One thread per (ray, 64-sample chunk).
// Set-to-1 scatter is idempotent -> plain byte stores, no atomics.
// voxel = round(A_inv @ world + t_inv); world = src + s * stepvec.
// ---------------------------------------------------------------------------
__global__ __launch_bounds__(256) void bp_scatter(
    const float* __restrict__ predF, const float* __restrict__ predL,
    const float* __restrict__ srcF,  const float* __restrict__ tgtF,
    const float* __restrict__ srcL,  const float* __restrict__ tgtL,
    const float* __restrict__ Ainv,  const float* __restrict__ tinv,
    unsigned char* __restrict__ vols)
{
  int tid   = blockIdx.x * blockDim.x + threadIdx.x;
  int ray   = tid >> 3;                 // 8 chunks x 64 samples = 512 samples
  int chunk = tid & 7;
  int p     = ray & (DETN * DETN - 1);  // detector pixel
  int bv    = ray >> 14;                // b*2 + view, in [0, NB*2)
  int view  = bv & 1;
  int b     = bv >> 1;

  const float* pred = view ? predL : predF;
  if (pred[(b << 14) + p] <= 0.5f) return;   // inactive ray: contributes nothing

  const float* src = view ? srcL : srcF;
  const float* tg  = (view ? tgtL : tgtF) + p * 3;

  float sx = src[0], sy = src[1], sz = src[2];
  float dx = tg[0] - sx, dy = tg[1] - sy, dz = tg[2] - sz;
  float len = sqrtf(dx*dx + dy*dy + dz*dz);
  // t_s = (s/(S-1)) * len * 2.5 along dir/(len+1e-8)
  float sc  = (len / (len + 1e-8f)) * (2.5f / (float)(NSAMP - 1));
  float ux = dx * sc, uy = dy * sc, uz = dz * sc;

  float m00=Ainv[0], m01=Ainv[1], m02=Ainv[2];
  float m10=Ainv[3], m11=Ainv[4], m12=Ainv[5];
  float m20=Ainv[6], m21=Ainv[7], m22=Ainv[8];

  // voxel-space base point and per-sample step (1 FMA per axis per sample)
  float bx = m00*sx + m01*sy + m02*sz + tinv[0];
  float by = m10*sx + m11*sy + m12*sz + tinv[1];
  float bz = m20*sx + m21*sy + m22*sz + tinv[2];
  float tx = m00*ux + m01*uy + m02*uz;
  float ty = m10*ux + m11*uy + m12*uz;
  float tz = m20*ux + m21*uy + m22*uz;

  unsigned char* vol = vols + ((size_t)bv << 21);
  int s0 = chunk << 6;
  #pragma unroll 4
  for (int s = s0; s < s0 + 64; ++s) {
    float fs = (float)s;
    int vx = (int)rintf(fmaf(fs, tx, bx));   // rintf == jnp.round (RNE)
    int vy = (int)rintf(fmaf(fs, ty, by));
    int vz = (int)rintf(fmaf(fs, tz, bz));
    if (((unsigned)vx < (unsigned)VDIM) &
        ((unsigned)vy < (unsigned)VDIM) &
        ((unsigned)vz < (unsigned)VDIM)) {
      vol[(((vx << 7) + vy) << 7) + vz] = (unsigned char)1;
    }
  }
}

// ---------------------------------------------------------------------------
// Kernel 2: BCE reduction. Per-voxel term (per batch): g*s + Q(s), with Q the
// quadratic that is exact on s in {0,1,2} for log1p(-sigmoid(s)), using
// logit(sigmoid(s)) == s. gt (f32) is double-buffered into LDS with
// GLOBAL_LOAD_ASYNC_TO_LDS_B128 (ASYNCcnt, in-order: wait<=1 => current tile
// resident while next is in flight); each lane reads back only the 16B its
// own async op wrote, so no barrier is needed. The 4 f-values per lane are
// folded into two v_wmma_f32_16x16x4_f32 ops against an all-ones B (64 f32
// terms per WMMA on the matrix pipe, co-executing with the VALU unpack math).
// Every D column replicates the row-sum -> divide block sum by 16.
// ---------------------------------------------------------------------------
__global__ __launch_bounds__(256) void bce_reduce(
    const float* __restrict__ gt, const unsigned char* __restrict__ vols,
    float* __restrict__ out)
{
  const float B0c = -0.69314718056f;   // log1p(-sigmoid(0))
  const float B1c = -0.52333859860f;   // quadratic-fit linear coeff
  const float B2c = -0.09677590830f;   // quadratic-fit square coeff

  int gtid = blockIdx.x * blockDim.x + threadIdx.x;
  int step = gridDim.x * blockDim.x * 4;   // voxels consumed per round

  const unsigned char* vA = vols;                    // b0 frontal
  const unsigned char* vB = vols + (1u << 21);       // b0 lateral
  const unsigned char* vC = vols + (2u << 21);       // b1 frontal
  const unsigned char* vD = vols + (3u << 21);       // b1 lateral

  v8f c = {};                 // 16x16 f32 accumulator (8 VGPRs, wave32)
  v2f ones; ones[0] = 1.0f; ones[1] = 1.0f;

#if HAVE_ASYNC_LDS
  __shared__ float4 gbuf[2][256];      // 8KB: 16B private slot per thread/buf
  __builtin_amdgcn_global_load_async_to_lds_b128(
      (g_v4i_p)(gt + gtid * 4),
      (l_v4i_p)&gbuf[0][threadIdx.x], 0, 0);
  int buf = 0;
#endif

  for (int i = gtid * 4; i < NVOX; i += step) {
#if HAVE_ASYNC_LDS
    // Issue next tile (dummy in-bounds address on the last round so exactly
    // two requests are always outstanding -> wait<=1 == "current tile done").
    int inext = i + step;
    const float* nsrc = (inext < NVOX) ? (gt + inext) : gt;
    __builtin_amdgcn_global_load_async_to_lds_b128(
        (g_v4i_p)nsrc,
        (l_v4i_p)&gbuf[buf ^ 1][threadIdx.x], 0, 0);
    __builtin_amdgcn_s_wait_asynccnt(1);
    __asm__ volatile("" ::: "memory");   // keep LDS read below the wait
    float4 g4 = gbuf[buf][threadIdx.x];
    buf ^= 1;
#else
    float4 g4 = *(const float4*)(gt + i);
    __builtin_prefetch(gt + i + step, 0, 1);
#endif
    unsigned wA = *(const unsigned*)(vA + i);
    unsigned wB = *(const unsigned*)(vB + i);
    unsigned wC = *(const unsigned*)(vC + i);
    unsigned wD = *(const unsigned*)(vD + i);
    __builtin_prefetch(vA + i + step, 0, 1);   // global_prefetch_b8
    __builtin_prefetch(vC + i + step, 0, 1);

    float gk[4] = {g4.x, g4.y, g4.z, g4.w};
    float f[4];
    #pragma unroll
    for (int k = 0; k < 4; ++k) {
      float s0 = (float)(((wA >> (8*k)) & 0xFFu) + ((wB >> (8*k)) & 0xFFu));
      float s1 = (float)(((wC >> (8*k)) & 0xFFu) + ((wD >> (8*k)) & 0xFFu));
      float gb1 = gk[k] + B1c;
      f[k] = fmaf(s0, fmaf(B2c, s0, gb1), B0c)
           + fmaf(s1, fmaf(B2c, s1, gb1), B0c);
    }
    v2f a0; a0[0] = f[0]; a0[1] = f[1];
    v2f a1; a1[0] = f[2]; a1[1] = f[3];
    // D = A x ones + C : folds 64 f32 terms per instruction on the matrix pipe
    c = __builtin_amdgcn_wmma_f32_16x16x4_f32(false, a0, false, ones,
                                              (short)0, c, false, false);
    c = __builtin_amdgcn_wmma_f32_16x16x4_f32(false, a1, false, ones,
                                              (short)0, c, false, false);
  }

  float s = c[0]+c[1]+c[2]+c[3]+c[4]+c[5]+c[6]+c[7];

  __shared__ float red[256];
  red[threadIdx.x] = s;
  __syncthreads();
  for (int off = 128; off > 0; off >>= 1) {
    if ((int)threadIdx.x < off) red[threadIdx.x] += red[threadIdx.x + off];
    __syncthreads();
  }
  if (threadIdx.x == 0) {
    // /16: ones-B replicates each row-sum across 16 D columns
    const float SCALE = -1.0f / (16.0f * (float)NVOX * (float)NB);
    atomicAdd(out, red[0] * SCALE);
  }
}

// ---------------------------------------------------------------------------
extern "C" void kernel_launch(void* const* d_in, const int* in_sizes, int n_in,
                              void* d_out, int out_size, void* d_ws, size_t ws_size,
                              hipStream_t stream) {
  const float* predF = (const float*)d_in[0];
  const float* predL = (const float*)d_in[1];
  const float* srcF  = (const float*)d_in[2];
  const float* tgtF  = (const float*)d_in[3];
  const float* srcL  = (const float*)d_in[4];
  const float* tgtL  = (const float*)d_in[5];
  const float* gt    = (const float*)d_in[6];
  const float* Ainv  = (const float*)d_in[7];
  const float* tinv  = (const float*)d_in[8];
  float* out = (float*)d_out;
  unsigned char* vols = (unsigned char*)d_ws;

  (void)in_sizes; (void)n_in; (void)out_size; (void)ws_size;

  // 8 MB of volumes as 524288 uint4 words
  zero_init<<<512, 256, 0, stream>>>((uint4*)vols, out,
                                     (int)(((size_t)NVOLS * NVOX) / 16));

  // NB*2*128*128 rays x 8 chunks = 524288 threads
  bp_scatter<<<(NB * 2 * DETN * DETN * 8) / 256, 256, 0, stream>>>(
      predF, predL, srcF, tgtF, srcL, tgtL, Ainv, tinv, vols);

  bce_reduce<<<1024, 256, 0, stream>>>(gt, vols, out);
}